// _CRLI_62208306315774
// MI455X (gfx1250) — compile-verified
//
#include <hip/hip_runtime.h>
#include <math.h>

#define BZ 2048
#define NS 256
#define NF 64
#define HH 64
#define LAMBDA 0.01f

typedef __attribute__((ext_vector_type(2))) float v2f;
typedef __attribute__((ext_vector_type(8))) float v8f;

// ---------------------------------------------------------------------------
// Optional CDNA5 async global->LDS path (ASYNCcnt-tracked DMA-style copies).
// Builtin signature (from hipcc diagnostic): param0 = addrspace(1) v4i*,
// param1 = addrspace(3) v4i*, then two constant ints (offset, cpol).
// ---------------------------------------------------------------------------
#if __has_builtin(__builtin_amdgcn_global_load_async_to_lds_b128) && \
    __has_builtin(__builtin_amdgcn_s_wait_asynccnt)
#define CRLI_ASYNC 1
typedef int v4i_gcc __attribute__((vector_size(16)));
typedef __attribute__((address_space(1))) v4i_gcc gas_v4i;
typedef __attribute__((address_space(3))) v4i_gcc las_v4i;
__device__ __forceinline__ void async_cp16(const float* gsrc, float* ldst) {
  gas_v4i* gp = (gas_v4i*)(unsigned long long)(uintptr_t)gsrc;
  las_v4i* lp = (las_v4i*)(unsigned int)(uintptr_t)ldst;
  __builtin_amdgcn_global_load_async_to_lds_b128(gp, lp, 0, 0);
}
#endif

// Prefetch a 16x64 f32 tile (rows row0..row0+15, fixed t, row stride NS*64)
// into LDS. Async if available, else synchronous vector copy.
__device__ __forceinline__ void xtile_prefetch(const float* __restrict__ src,
                                               size_t row0, int t, float* dst) {
  const int lane = threadIdx.x & 31;
#ifdef CRLI_ASYNC
#pragma unroll
  for (int i = 0; i < 8; ++i) {
    int idx = i * 32 + lane;            // 256 groups of 4 floats (16B)
    int row = idx >> 4, cg = idx & 15;
    const float* g = src + ((row0 + row) * NS + t) * NF + cg * 4;
    async_cp16(g, dst + idx * 4);
  }
#else
  for (int i = lane; i < 16 * 64; i += 32) {
    int row = i >> 6, c = i & 63;
    dst[i] = src[((row0 + row) * NS + t) * NF + c];
  }
#endif
}

// Wait until the tile we are about to consume is resident in LDS.
// pending==1: one newer prefetch may remain in flight (in-order completion).
__device__ __forceinline__ void xtile_wait(int pending) {
#ifdef CRLI_ASYNC
  if (pending) __builtin_amdgcn_s_wait_asynccnt(1);
  else         __builtin_amdgcn_s_wait_asynccnt(0);
#else
  (void)pending;
#endif
}

// ---------------------------------------------------------------------------
// WMMA f32 16x16x4 micro-kernel.  D = A(16x4) * B(4x16) + C.
// A layout: lane&15 = M row; K pair selected by lane>>4.
// B layout: lane&15 = N col; same K pairing.
// ---------------------------------------------------------------------------
__device__ __forceinline__ v8f wmma4(v2f a, v2f b, v8f c) {
  return __builtin_amdgcn_wmma_f32_16x16x4_f32(false, a, false, b, (short)0, c,
                                               false, false);
}

// acc += A(16 x K, row-major lda, LDS) * W(n0..n0+15 rows of an N x K
// row-major weight matrix, i.e. B = W^T).  Rows >= Nvalid contribute zero.
// Fast unguarded path when the tile is fully inside the matrix.
__device__ __forceinline__ v8f gemm_tile(const float* A, int lda,
                                         const float* __restrict__ W, int ldw,
                                         int n0, int Nvalid, int K, v8f acc) {
  const int lane = threadIdx.x & 31;
  const int r    = lane & 15;
  const int half = lane >> 4;
  const int n    = n0 + r;
  if (n0 + 16 <= Nvalid) {
    const float* wrow = W + (size_t)n * ldw;
#pragma unroll
    for (int k = 0; k < K; k += 4) {
      v2f a, b;
      a.x = A[r * lda + k + 2 * half];
      a.y = A[r * lda + k + 2 * half + 1];
      b.x = wrow[k + 2 * half];
      b.y = wrow[k + 2 * half + 1];
      acc = wmma4(a, b, acc);
    }
  } else {
    const bool nv = (n < Nvalid);
    const float* wrow = W + (size_t)(nv ? n : 0) * ldw;
#pragma unroll
    for (int k = 0; k < K; k += 4) {
      v2f a, b;
      a.x = A[r * lda + k + 2 * half];
      a.y = A[r * lda + k + 2 * half + 1];
      b.x = nv ? wrow[k + 2 * half]     : 0.0f;
      b.y = nv ? wrow[k + 2 * half + 1] : 0.0f;
      acc = wmma4(a, b, acc);
    }
  }
  return acc;
}

// D tile (16x16) -> LDS.  VGPR j holds row (j + 8*(lane>>4)), col = lane&15.
__device__ __forceinline__ void store_tile(float* D, int ldd, int n0, v8f acc) {
  const int lane = threadIdx.x & 31;
  const int c    = lane & 15;
  const int half = lane >> 4;
#pragma unroll
  for (int j = 0; j < 8; ++j) D[(j + 8 * half) * ldd + n0 + c] = acc[j];
}

__device__ __forceinline__ float wave_sum(float v) {
#pragma unroll
  for (int i = 16; i > 0; i >>= 1) v += __shfl_xor(v, i, 32);
  return v;
}

__device__ __forceinline__ float sigmoidf_(float x) {
  return 1.0f / (1.0f + __expf(-x));
}

// ---------------------------------------------------------------------------
// Generic one-wave GRU cell, templated on dims so every loop unrolls and the
// elementwise index math becomes shifts.  x may alias h (self-feeding GRU).
// x: 16 x DIN (ld = DIN), h: 16 x DH (in place), gi/gh: 16 x ceil16(3*DH).
// ---------------------------------------------------------------------------
struct GruW { const float *Wih, *Whh, *bih, *bhh; };

template <int DIN, int DH>
__device__ void gru_cell_wave(const GruW& P, const float* x, float* h,
                              float* gi, float* gh) {
  constexpr int G3  = 3 * DH;
  constexpr int NT  = (G3 + 15) / 16;
  constexpr int LDG = NT * 16;
  const int lane = threadIdx.x & 31;
#pragma unroll
  for (int nt = 0; nt < NT; ++nt) {
    v8f ai = {}, ah = {};
    ai = gemm_tile(x, DIN, P.Wih, DIN, nt * 16, G3, DIN, ai);
    ah = gemm_tile(h, DH,  P.Whh, DH,  nt * 16, G3, DH,  ah);
    store_tile(gi, LDG, nt * 16, ai);
    store_tile(gh, LDG, nt * 16, ah);
  }
  __syncthreads();
  for (int idx = lane; idx < 16 * DH; idx += 32) {
    int m = idx / DH, c = idx % DH;       // DH is a power of two -> shifts
    float ir  = gi[m * LDG + c]          + P.bih[c];
    float iz  = gi[m * LDG + DH + c]     + P.bih[DH + c];
    float in_ = gi[m * LDG + 2 * DH + c] + P.bih[2 * DH + c];
    float hr  = gh[m * LDG + c]          + P.bhh[c];
    float hz  = gh[m * LDG + DH + c]     + P.bhh[DH + c];
    float hn  = gh[m * LDG + 2 * DH + c] + P.bhh[2 * DH + c];
    float r  = sigmoidf_(ir + hr);
    float z  = sigmoidf_(iz + hz);
    float nn = tanhf(in_ + r * hn);
    h[m * DH + c] = (1.0f - z) * nn + z * h[m * DH + c];
  }
  __syncthreads();
}

struct Mrnn2 { GruW c0, c1; };
struct DiscP { GruW c0, c1, c2, c3, c4; const float *Wo, *bo; };

// ---------------------------------------------------------------------------
// K0: zero the loss accumulators.
// ---------------------------------------------------------------------------
__global__ void k_zero(float* a) {
  if (threadIdx.x < 8) a[threadIdx.x] = 0.0f;
}

// ---------------------------------------------------------------------------
// K1: fused forward/backward 2-layer GRU over the full sequence.
// grid (BZ/16, 2): x = batch tile, y = direction (param-set select).
// Async double-buffered x tile overlaps HBM/L2 latency with WMMA compute.
// ---------------------------------------------------------------------------
__global__ __launch_bounds__(32)
void k_mrnn(const float* __restrict__ X, Mrnn2 Pf, Mrnn2 Pb,
            float* f_hs, float* b_hs, float* fb) {
  __shared__ __attribute__((aligned(16))) float sx[2][16 * 64];
  __shared__ float sh[16 * 64], sgi[16 * 192], sgh[16 * 192];
  const size_t b0 = (size_t)blockIdx.x * 16;
  const int dir   = blockIdx.y;
  const Mrnn2 P   = dir ? Pb : Pf;
  float* hs       = dir ? b_hs : f_hs;
  const int lane  = threadIdx.x;
  for (int i = lane; i < 16 * 64; i += 32) sh[i] = 0.0f;
  xtile_prefetch(X, b0, 0, sx[0]);
  __syncthreads();
  for (int t = 0; t < NS; ++t) {
    const int pend = (t + 1 < NS);
    if (pend) xtile_prefetch(X, b0, t + 1, sx[(t + 1) & 1]);
    xtile_wait(pend);
    __syncthreads();
    gru_cell_wave<64, 64>(P.c0, sx[t & 1], sh, sgi, sgh);
    gru_cell_wave<64, 64>(P.c1, sh, sh, sgi, sgh);
    for (int i = lane; i < 16 * 64; i += 32)
      hs[((b0 + (i >> 6)) * NS + t) * HH + (i & 63)] = sh[i];
  }
  __syncthreads();
  for (int i = lane; i < 16 * 64; i += 32)
    fb[(b0 + (i >> 6)) * 128 + dir * 64 + (i & 63)] = sh[i];
}

// ---------------------------------------------------------------------------
// K2: projections (H->NF), bidirectional combine, imputation, l_pre terms.
// grid = BZ*NS/16 waves, each owns 16 consecutive t for one batch row group.
// ---------------------------------------------------------------------------
__global__ __launch_bounds__(32)
void k_impute(const float* __restrict__ X, const float* __restrict__ mask,
              const float* __restrict__ f_hs, const float* __restrict__ b_hs,
              const float* WoF, const float* boF,
              const float* WoB, const float* boB,
              float* imp_out, float* accum) {
  __shared__ float sA[16 * 64], sF[16 * 64], sB[16 * 64];
  const int tile = blockIdx.x;
  const int b    = tile >> 4;
  const int t0   = (tile & 15) * 16;
  const int lane = threadIdx.x;
  for (int i = lane; i < 16 * 64; i += 32)
    sA[i] = f_hs[(((size_t)b) * NS + t0 + (i >> 6)) * HH + (i & 63)];
  __syncthreads();
#pragma unroll
  for (int nt = 0; nt < 4; ++nt) {
    v8f a = {};
    a = gemm_tile(sA, 64, WoF, 64, nt * 16, 64, 64, a);
    store_tile(sF, 64, nt * 16, a);
  }
  __syncthreads();
  for (int i = lane; i < 16 * 64; i += 32)
    sA[i] = b_hs[(((size_t)b) * NS + (NS - 1 - (t0 + (i >> 6)))) * HH + (i & 63)];
  __syncthreads();
#pragma unroll
  for (int nt = 0; nt < 4; ++nt) {
    v8f a = {};
    a = gemm_tile(sA, 64, WoB, 64, nt * 16, 64, 64, a);
    store_tile(sB, 64, nt * 16, a);
  }
  __syncthreads();
  float pre = 0.0f, msum = 0.0f;
  for (int i = lane; i < 16 * 64; i += 32) {
    int m = i >> 6, c = i & 63;
    size_t g = (((size_t)b) * NS + t0 + m) * NF + c;
    float imp = 0.5f * ((sF[i] + boF[c]) + (sB[i] + boB[c]));
    float x = X[g], mk = mask[g];
    imp_out[g] = x * mk + imp * (1.0f - mk);
    float d = imp - x;
    pre += mk * d * d;
    msum += mk;
  }
  pre = wave_sum(pre); msum = wave_sum(msum);
  if (lane == 0) { atomicAdd(&accum[0], pre); atomicAdd(&accum[1], msum); }
}

// ---------------------------------------------------------------------------
// K3: 5-cell discriminator stack + projection + l_G accumulation.
// Same async double-buffered input tile as K1.
// ---------------------------------------------------------------------------
__global__ __launch_bounds__(32)
void k_disc(const float* __restrict__ imp, const float* __restrict__ mask,
            DiscP P, float* accum) {
  __shared__ __attribute__((aligned(16))) float sx[2][16 * 64];
  __shared__ float h0[16 * 32], h1[16 * 16], h2[16 * 8], h3[16 * 16], h4[16 * 32];
  __shared__ float sgi[16 * 96], sgh[16 * 96];
  const size_t b0 = (size_t)blockIdx.x * 16;
  const int lane  = threadIdx.x;
  for (int i = lane; i < 16 * 32; i += 32) { h0[i] = 0.0f; h4[i] = 0.0f; }
  for (int i = lane; i < 16 * 16; i += 32) { h1[i] = 0.0f; h3[i] = 0.0f; }
  for (int i = lane; i < 16 * 8;  i += 32) h2[i] = 0.0f;
  xtile_prefetch(imp, b0, 0, sx[0]);
  __syncthreads();
  float g = 0.0f;
  for (int t = 0; t < NS; ++t) {
    const int pend = (t + 1 < NS);
    if (pend) xtile_prefetch(imp, b0, t + 1, sx[(t + 1) & 1]);
    xtile_wait(pend);
    __syncthreads();
    gru_cell_wave<64, 32>(P.c0, sx[t & 1], h0, sgi, sgh);
    gru_cell_wave<32, 16>(P.c1, h0, h1, sgi, sgh);
    gru_cell_wave<16, 8>(P.c2, h1, h2, sgi, sgh);
    gru_cell_wave<8, 16>(P.c3, h2, h3, sgi, sgh);
    gru_cell_wave<16, 32>(P.c4, h3, h4, sgi, sgh);
#pragma unroll
    for (int nt = 0; nt < 4; ++nt) {
      v8f a = {};
      a = gemm_tile(h4, 32, P.Wo, 32, nt * 16, 64, 32, a);
      int c = nt * 16 + (lane & 15), half = lane >> 4;
#pragma unroll
      for (int j = 0; j < 8; ++j) {
        int m = j + 8 * half;
        size_t gi = ((b0 + m) * NS + t) * NF + c;
        float d  = a[j] + P.bo[c];
        float w  = 1.0f - mask[gi];
        float sp = d > 0.0f ? d + log1pf(__expf(-d)) : log1pf(__expf(d));
        g += w * (sp - d * w);
      }
    }
    __syncthreads();
  }
  g = wave_sum(g);
  if (lane == 0) atomicAdd(&accum[3], g);
}

// ---------------------------------------------------------------------------
// K4: decoder FCN (128->128->64), stores z, accumulates tr(H^T H).
// ---------------------------------------------------------------------------
__global__ __launch_bounds__(32)
void k_fcn(const float* __restrict__ fb, const float* W0, const float* b0v,
           const float* W1, const float* b1v, float* z, float* accum) {
  __shared__ float sin_[16 * 128], smid[16 * 128];
  const int bb   = blockIdx.x * 16;
  const int lane = threadIdx.x;
  for (int i = lane; i < 16 * 128; i += 32)
    sin_[i] = fb[(size_t)(bb + (i >> 7)) * 128 + (i & 127)];
  __syncthreads();
#pragma unroll
  for (int nt = 0; nt < 8; ++nt) {
    v8f a = {};
    a = gemm_tile(sin_, 128, W0, 128, nt * 16, 128, 128, a);
    store_tile(smid, 128, nt * 16, a);
  }
  __syncthreads();
  for (int i = lane; i < 16 * 128; i += 32) smid[i] += b0v[i & 127];
  __syncthreads();
  float tr = 0.0f;
#pragma unroll
  for (int nt = 0; nt < 4; ++nt) {
    v8f a = {};
    a = gemm_tile(smid, 128, W1, 128, nt * 16, 64, 128, a);
    int c = nt * 16 + (lane & 15), half = lane >> 4;
#pragma unroll
    for (int j = 0; j < 8; ++j) {
      int m = j + 8 * half;
      float v = a[j] + b1v[c];
      z[(size_t)(bb + m) * 64 + c] = v;
      tr += v * v;
    }
  }
  tr = wave_sum(tr);
  if (lane == 0) atomicAdd(&accum[4], tr);
}

// ---------------------------------------------------------------------------
// K5: decoder self-feeding GRU + projection + l_rec accumulation (no store).
// ---------------------------------------------------------------------------
__global__ __launch_bounds__(32)
void k_dec(const float* __restrict__ X, const float* __restrict__ mask,
           const float* __restrict__ z, GruW g, const float* Wo,
           const float* bo, float* accum) {
  __shared__ float sh[16 * 64], sgi[16 * 192], sgh[16 * 192];
  const size_t b0 = (size_t)blockIdx.x * 16;
  const int lane  = threadIdx.x;
  for (int i = lane; i < 16 * 64; i += 32)
    sh[i] = z[(b0 + (i >> 6)) * 64 + (i & 63)];
  __syncthreads();
  float rec = 0.0f;
  for (int t = 0; t < NS; ++t) {
    gru_cell_wave<64, 64>(g, sh, sh, sgi, sgh);
#pragma unroll
    for (int nt = 0; nt < 4; ++nt) {
      v8f a = {};
      a = gemm_tile(sh, 64, Wo, 64, nt * 16, 64, 64, a);
      int c = nt * 16 + (lane & 15), half = lane >> 4;
#pragma unroll
      for (int j = 0; j < 8; ++j) {
        int m = j + 8 * half;
        size_t gi = ((b0 + m) * NS + t) * NF + c;
        float d = a[j] + bo[c] - X[gi];
        rec += mask[gi] * d * d;
      }
    }
  }
  rec = wave_sum(rec);
  if (lane == 0) atomicAdd(&accum[2], rec);
}

// ---------------------------------------------------------------------------
// K6: L^T F = z^T (64 x 2048) @ F (2048 x 8); accumulate ||L^T F||^2.
// grid = 4 waves, one per 16-column tile of z.
// ---------------------------------------------------------------------------
__global__ __launch_bounds__(32)
void k_kmeans(const float* __restrict__ z, const float* __restrict__ F,
              float* accum) {
  const int m0   = blockIdx.x * 16;
  const int lane = threadIdx.x;
  const int r    = lane & 15;
  const int half = lane >> 4;
  v8f acc = {};
  for (int k = 0; k < BZ; k += 4) {
    v2f a, b;
    a.x = z[(size_t)(k + 2 * half) * 64 + m0 + r];
    a.y = z[(size_t)(k + 2 * half + 1) * 64 + m0 + r];
    b.x = (r < 8) ? F[(size_t)(k + 2 * half) * 8 + r] : 0.0f;
    b.y = (r < 8) ? F[(size_t)(k + 2 * half + 1) * 8 + r] : 0.0f;
    acc = wmma4(a, b, acc);
  }
  float p = 0.0f;
  if (r < 8) {
#pragma unroll
    for (int j = 0; j < 8; ++j) p += acc[j] * acc[j];
  }
  p = wave_sum(p);
  if (lane == 0) atomicAdd(&accum[5], p);
}

// ---------------------------------------------------------------------------
// K7: combine losses.
// ---------------------------------------------------------------------------
__global__ void k_final(const float* a, float* out) {
  if (threadIdx.x == 0 && blockIdx.x == 0) {
    float lG   = a[3] / (float)((size_t)BZ * NS * NF);
    float lpre = a[0] / (a[1] + 1e-12f);
    float lrec = a[2] / (a[1] + 1e-12f);
    out[0] = lG + lpre + lrec + LAMBDA * (a[4] - a[5]);
  }
}

// ---------------------------------------------------------------------------
// Host launcher.  Input order = jax pytree flatten of setup_inputs():
// X, missing_mask, term_F, then params sorted by key:
//  b.Wo, b.bo, b.cells[0](4), b.cells[1](4),
//  dec.Wo, dec.bo, dec.fcn[0](2), dec.fcn[1](2), dec.gru(4),
//  disc.Wo, disc.bo, disc.cells[0..4](4 each),
//  f.Wo, f.bo, f.cells[0](4), f.cells[1](4)
// ---------------------------------------------------------------------------
static inline const float* fp(void* const* d_in, int i) {
  return (const float*)d_in[i];
}

extern "C" void kernel_launch(void* const* d_in, const int* in_sizes, int n_in,
                              void* d_out, int out_size, void* d_ws,
                              size_t ws_size, hipStream_t stream) {
  (void)in_sizes; (void)n_in; (void)out_size; (void)ws_size;
  const float* X     = fp(d_in, 0);
  const float* mask  = fp(d_in, 1);
  const float* termF = fp(d_in, 2);

  const float* bWo = fp(d_in, 3);
  const float* bbo = fp(d_in, 4);
  Mrnn2 Pb = { { fp(d_in, 5),  fp(d_in, 6),  fp(d_in, 7),  fp(d_in, 8)  },
               { fp(d_in, 9),  fp(d_in, 10), fp(d_in, 11), fp(d_in, 12) } };

  const float* decWo = fp(d_in, 13);
  const float* decbo = fp(d_in, 14);
  const float* fcW0  = fp(d_in, 15);
  const float* fcb0  = fp(d_in, 16);
  const float* fcW1  = fp(d_in, 17);
  const float* fcb1  = fp(d_in, 18);
  GruW decG = { fp(d_in, 19), fp(d_in, 20), fp(d_in, 21), fp(d_in, 22) };

  DiscP DP;
  DP.Wo = fp(d_in, 23);
  DP.bo = fp(d_in, 24);
  DP.c0 = { fp(d_in, 25), fp(d_in, 26), fp(d_in, 27), fp(d_in, 28) };
  DP.c1 = { fp(d_in, 29), fp(d_in, 30), fp(d_in, 31), fp(d_in, 32) };
  DP.c2 = { fp(d_in, 33), fp(d_in, 34), fp(d_in, 35), fp(d_in, 36) };
  DP.c3 = { fp(d_in, 37), fp(d_in, 38), fp(d_in, 39), fp(d_in, 40) };
  DP.c4 = { fp(d_in, 41), fp(d_in, 42), fp(d_in, 43), fp(d_in, 44) };

  const float* fWo = fp(d_in, 45);
  const float* fbo = fp(d_in, 46);
  Mrnn2 Pf = { { fp(d_in, 47), fp(d_in, 48), fp(d_in, 49), fp(d_in, 50) },
               { fp(d_in, 51), fp(d_in, 52), fp(d_in, 53), fp(d_in, 54) } };

  // workspace layout (floats)
  const size_t seqN = (size_t)BZ * NS * NF;
  float* w     = (float*)d_ws;
  float* f_hs  = w;
  float* b_hs  = f_hs + seqN;
  float* imp   = b_hs + seqN;
  float* fbuf  = imp + seqN;
  float* zbuf  = fbuf + (size_t)BZ * 128;
  float* accum = zbuf + (size_t)BZ * 64;

  k_zero<<<1, 32, 0, stream>>>(accum);
  k_mrnn<<<dim3(BZ / 16, 2), 32, 0, stream>>>(X, Pf, Pb, f_hs, b_hs, fbuf);
  k_impute<<<BZ * NS / 16, 32, 0, stream>>>(X, mask, f_hs, b_hs, fWo, fbo,
                                            bWo, bbo, imp, accum);
  k_disc<<<BZ / 16, 32, 0, stream>>>(imp, mask, DP, accum);
  k_fcn<<<BZ / 16, 32, 0, stream>>>(fbuf, fcW0, fcb0, fcW1, fcb1, zbuf, accum);
  k_dec<<<BZ / 16, 32, 0, stream>>>(X, mask, zbuf, decG, decWo, decbo, accum);
  k_kmeans<<<4, 32, 0, stream>>>(zbuf, termF, accum);
  k_final<<<1, 32, 0, stream>>>(accum, (float*)d_out);
}